// GNNActor_42210938585671
// MI455X (gfx1250) — compile-verified
//
#include <hip/hip_runtime.h>
#include <math.h>

// ---------------------------------------------------------------------------
// GNNActor pipeline for gfx1250 (MI455X).
// Inputs (dict order):
//  0 x[N,128] f32        1 edge_index[2,E] int   2 batch[N] int
//  3 conv_w[128,128]     4 conv_b[128]
//  5 lin1_w[32,128]      6 lin1_b[32]
//  7 lin2_w[32,32]       8 lin2_b[32]
//  9 dc_w[1,32]         10 dc_b[1]
// 11 mu_w[1,32]         12 mu_b[1]
// 13 std_w[1,32]        14 std_b[1]
// Output: concat( dirichlet[N], mu[G], sigma[G] ), G = 128.
// ---------------------------------------------------------------------------

typedef __attribute__((ext_vector_type(2))) float v2f;
typedef __attribute__((ext_vector_type(8))) float v8f;

#define GRAPHS 128

// ---------------------------------------------------------------------------
// init: zero agg [N*128], pool sums/counts; deg = 1.0 (self loop)
// ---------------------------------------------------------------------------
__global__ void init_ws_kernel(float* __restrict__ agg, float* __restrict__ deg,
                               float* __restrict__ psum, float* __restrict__ pcnt,
                               long total, int n_nodes) {
    long i = (long)blockIdx.x * blockDim.x + threadIdx.x;
    if (i < total) agg[i] = 0.0f;
    if (i < n_nodes) deg[i] = 1.0f;
    if (i < GRAPHS * 32) psum[i] = 0.0f;
    if (i < GRAPHS) pcnt[i] = 0.0f;
}

// ---------------------------------------------------------------------------
// degree: deg[dst] += 1 per edge
// ---------------------------------------------------------------------------
__global__ void degree_kernel(const int* __restrict__ dst, float* __restrict__ deg, int E) {
    int e = blockIdx.x * blockDim.x + threadIdx.x;
    if (e < E) atomicAdd(&deg[dst[e]], 1.0f);
}

__global__ void dinv_kernel(const float* __restrict__ deg, float* __restrict__ dinv, int n) {
    int i = blockIdx.x * blockDim.x + threadIdx.x;
    if (i < n) dinv[i] = rsqrtf(deg[i]);
}

// ---------------------------------------------------------------------------
// WMMA f32 GEMM:  out[N, NCOLS] = A[N, K] @ W[NCOLS, K]^T   (+ bias, relu)
// One wave per 16x16 output tile; K stepped by 4 via V_WMMA_F32_16X16X4_F32.
// A-operand layout (ISA 7.12.2, 32-bit 16x4): lanes 0-15 hold K=0 (v0), K=1 (v1);
// lanes 16-31 hold K=2,K=3. B mirrored. C/D: lane l holds col l, rows half*8+v.
// ---------------------------------------------------------------------------
template <int K, int NCOLS, bool BIAS_RELU>
__global__ void wmma_gemm_f32_kernel(const float* __restrict__ A,
                                     const float* __restrict__ W,
                                     const float* __restrict__ bias,
                                     float* __restrict__ out,
                                     int n_tile_rows) {
    const int wave = threadIdx.x >> 5;
    const int lane = threadIdx.x & 31;
    const int tilesPerRow = NCOLS / 16;
    const int tile = blockIdx.x * (blockDim.x >> 5) + wave;
    if (tile >= n_tile_rows * tilesPerRow) return;   // wave-uniform exit (EXEC all-1 inside)

    const int tr = tile / tilesPerRow;
    const int tc = tile % tilesPerRow;
    const int half = lane >> 4;
    const int l = lane & 15;

    const long row = (long)tr * 16 + l;       // A row this lane feeds
    const int col = tc * 16 + l;              // output column this lane owns in C/D

    const float* __restrict__ aRow = A + row * K;
    const float* __restrict__ wRow = W + (long)col * K;

    v8f acc = {};
#pragma unroll 4
    for (int k = 0; k < K; k += 4) {
        v2f a, b;
        a.x = aRow[k + 2 * half];
        a.y = aRow[k + 2 * half + 1];
        b.x = wRow[k + 2 * half];
        b.y = wRow[k + 2 * half + 1];
        acc = __builtin_amdgcn_wmma_f32_16x16x4_f32(false, a, false, b,
                                                    (short)0, acc, false, false);
    }

    float bv = 0.0f;
    if (BIAS_RELU) bv = bias[col];
#pragma unroll
    for (int v = 0; v < 8; ++v) {
        long r = (long)tr * 16 + half * 8 + v;
        float val = acc[v];
        if (BIAS_RELU) val = fmaxf(val + bv, 0.0f);
        out[r * NCOLS + col] = val;
    }
}

// ---------------------------------------------------------------------------
// edge scatter: agg[dst] += h[src] * dinv[src]*dinv[dst]
// one wave per edge, each lane owns a float4 chunk of the 128-dim row.
// h and agg both resident in the 192MB L2 -> L2-bound atomics.
// ---------------------------------------------------------------------------
__global__ void scatter_kernel(const float* __restrict__ h,
                               const int* __restrict__ src,
                               const int* __restrict__ dst,
                               const float* __restrict__ dinv,
                               float* __restrict__ agg, int E) {
    long t = (long)blockIdx.x * blockDim.x + threadIdx.x;
    int e = (int)(t >> 5);
    if (e >= E) return;
    int c = ((int)t & 31) * 4;
    int s = src[e];
    int d = dst[e];
    float norm = dinv[s] * dinv[d];
    const float4 hv = *(const float4*)&h[(long)s * 128 + c];
    float* ap = &agg[(long)d * 128 + c];
    atomicAdd(ap + 0, hv.x * norm);
    atomicAdd(ap + 1, hv.y * norm);
    atomicAdd(ap + 2, hv.z * norm);
    atomicAdd(ap + 3, hv.w * norm);
}

// ---------------------------------------------------------------------------
// finalize conv (in place into agg):
//   res = relu(agg + h*dinv^2 + conv_b) + x
// ---------------------------------------------------------------------------
__global__ void finalize_conv_kernel(const float* __restrict__ x,
                                     const float* __restrict__ h,
                                     float* __restrict__ agg,
                                     const float* __restrict__ conv_b,
                                     const float* __restrict__ dinv,
                                     long total) {
    long i = (long)blockIdx.x * blockDim.x + threadIdx.x;
    if (i >= total) return;
    int n = (int)(i >> 7);
    int d = (int)(i & 127);
    float di = dinv[n];
    float v = agg[i] + h[i] * di * di + conv_b[d];
    agg[i] = fmaxf(v, 0.0f) + x[i];
}

// ---------------------------------------------------------------------------
// per-node head: dirichlet = last @ dc_w^T + dc_b, pool accumulation
// ---------------------------------------------------------------------------
__global__ void node_reduce_kernel(const float* __restrict__ last,
                                   const int* __restrict__ batch,
                                   const float* __restrict__ dc_w,
                                   const float* __restrict__ dc_b,
                                   float* __restrict__ out_dc,
                                   float* __restrict__ psum,
                                   float* __restrict__ pcnt, int n_nodes) {
    int n = blockIdx.x * blockDim.x + threadIdx.x;
    if (n >= n_nodes) return;
    int b = batch[n];
    float acc = 0.0f;
    const float* row = last + (long)n * 32;
#pragma unroll
    for (int d = 0; d < 32; ++d) {
        float v = row[d];
        acc += v * dc_w[d];
        atomicAdd(&psum[b * 32 + d], v);
    }
    out_dc[n] = acc + dc_b[0];
    atomicAdd(&pcnt[b], 1.0f);
}

// ---------------------------------------------------------------------------
// per-graph head: mean pool, mu, sigma = softplus(...)
// ---------------------------------------------------------------------------
__global__ void pool_head_kernel(const float* __restrict__ psum,
                                 const float* __restrict__ pcnt,
                                 const float* __restrict__ mu_w,
                                 const float* __restrict__ mu_b,
                                 const float* __restrict__ std_w,
                                 const float* __restrict__ std_b,
                                 float* __restrict__ out_mu,
                                 float* __restrict__ out_sigma) {
    int g = threadIdx.x;
    if (g >= GRAPHS) return;
    float c = fmaxf(pcnt[g], 1.0f);
    float mu = 0.0f, sd = 0.0f;
#pragma unroll
    for (int d = 0; d < 32; ++d) {
        float p = psum[g * 32 + d] / c;
        mu += p * mu_w[d];
        sd += p * std_w[d];
    }
    out_mu[g] = mu + mu_b[0];
    float z = sd + std_b[0];
    out_sigma[g] = (z > 20.0f) ? z : log1pf(expf(z));   // stable softplus
}

// ---------------------------------------------------------------------------
extern "C" void kernel_launch(void* const* d_in, const int* in_sizes, int n_in,
                              void* d_out, int out_size, void* d_ws, size_t ws_size,
                              hipStream_t stream) {
    const float* x      = (const float*)d_in[0];
    const int*   ei     = (const int*)d_in[1];
    const int*   batch  = (const int*)d_in[2];
    const float* conv_w = (const float*)d_in[3];
    const float* conv_b = (const float*)d_in[4];
    const float* lin1_w = (const float*)d_in[5];
    const float* lin1_b = (const float*)d_in[6];
    const float* lin2_w = (const float*)d_in[7];
    const float* lin2_b = (const float*)d_in[8];
    const float* dc_w   = (const float*)d_in[9];
    const float* dc_b   = (const float*)d_in[10];
    const float* mu_w   = (const float*)d_in[11];
    const float* mu_b   = (const float*)d_in[12];
    const float* std_w  = (const float*)d_in[13];
    const float* std_b  = (const float*)d_in[14];

    const int D = 128;
    const int N = in_sizes[0] / D;       // 50000 (multiple of 16)
    const int E = in_sizes[1] / 2;       // 640000
    const int* src = ei;
    const int* dst = ei + E;

    // workspace carve-out (floats)
    float* ws   = (float*)d_ws;
    float* h    = ws;                          // N*128
    float* agg  = h    + (size_t)N * D;        // N*128 (becomes residual output)
    float* h1   = agg  + (size_t)N * D;        // N*32
    float* last = h1   + (size_t)N * 32;       // N*32
    float* deg  = last + (size_t)N * 32;       // N
    float* dinv = deg  + N;                    // N
    float* psum = dinv + N;                    // G*32
    float* pcnt = psum + GRAPHS * 32;          // G

    float* out_dc    = (float*)d_out;          // [N]
    float* out_mu    = out_dc + N;             // [G]
    float* out_sigma = out_mu + GRAPHS;        // [G]

    const long totalND = (long)N * D;
    const int  TPB = 256;

    // 1) init scratch (deg=1 self-loop, agg/pool zeroed) -- every call, graph-safe
    {
        long blocks = (totalND + TPB - 1) / TPB;
        init_ws_kernel<<<(int)blocks, TPB, 0, stream>>>(agg, deg, psum, pcnt, totalND, N);
    }
    // 2) degrees and dinv
    degree_kernel<<<(E + TPB - 1) / TPB, TPB, 0, stream>>>(dst, deg, E);
    dinv_kernel<<<(N + TPB - 1) / TPB, TPB, 0, stream>>>(deg, dinv, N);

    // 3) h = x @ conv_w^T   (WMMA f32, 3125x8 tiles)
    {
        int nTileRows = N / 16;
        int tiles = nTileRows * (D / 16);
        int blocks = (tiles + 7) / 8;          // 8 waves per 256-thread block
        wmma_gemm_f32_kernel<128, 128, false><<<blocks, TPB, 0, stream>>>(
            x, conv_w, nullptr, h, nTileRows);
    }

    // 4) edge aggregation (L2-resident gather + atomic scatter)
    {
        long threads = (long)E * 32;
        scatter_kernel<<<(int)((threads + TPB - 1) / TPB), TPB, 0, stream>>>(
            h, src, dst, dinv, agg, E);
    }

    // 5) fused conv epilogue + residual (in place into agg)
    finalize_conv_kernel<<<(int)((totalND + TPB - 1) / TPB), TPB, 0, stream>>>(
        x, h, agg, conv_b, dinv, totalND);

    // 6) h1 = relu(agg @ lin1_w^T + b1)   [N,128]x[128,32]
    {
        int nTileRows = N / 16;
        int tiles = nTileRows * (32 / 16);
        wmma_gemm_f32_kernel<128, 32, true><<<(tiles + 7) / 8, TPB, 0, stream>>>(
            agg, lin1_w, lin1_b, h1, nTileRows);
    }
    // 7) last = relu(h1 @ lin2_w^T + b2)  [N,32]x[32,32]
    {
        int nTileRows = N / 16;
        int tiles = nTileRows * (32 / 16);
        wmma_gemm_f32_kernel<32, 32, true><<<(tiles + 7) / 8, TPB, 0, stream>>>(
            h1, lin2_w, lin2_b, last, nTileRows);
    }

    // 8) dirichlet head + pool accumulation
    node_reduce_kernel<<<(N + TPB - 1) / TPB, TPB, 0, stream>>>(
        last, batch, dc_w, dc_b, out_dc, psum, pcnt, N);

    // 9) per-graph mu / sigma
    pool_head_kernel<<<1, GRAPHS, 0, stream>>>(
        psum, pcnt, mu_w, mu_b, std_w, std_b, out_mu, out_sigma);
}